// GRACE_37082747634687
// MI455X (gfx1250) — compile-verified
//
#include <hip/hip_runtime.h>
#include <hip/hip_bf16.h>

typedef __attribute__((ext_vector_type(2))) float v2f;
typedef __attribute__((ext_vector_type(8))) float v8f;

#define NODES 10000
#define EDGES 320000
#define INC   256
#define HID   512
#define OUTC  256
#define SCAN_T 256

// ---------------------------------------------------------------------------
// Zero fill (int)
// ---------------------------------------------------------------------------
__global__ __launch_bounds__(256) void GRACE_zero_i32(int* __restrict__ p, int n) {
    int i = blockIdx.x * blockDim.x + threadIdx.x;
    if (i < n) p[i] = 0;
}

// ---------------------------------------------------------------------------
// Degree count (int): deg[dst[e]] += 1
// ---------------------------------------------------------------------------
__global__ __launch_bounds__(256) void GRACE_count(const long long* __restrict__ dst,
                                                   int* __restrict__ deg, int nedges) {
    int e = blockIdx.x * blockDim.x + threadIdx.x;
    if (e < nedges) atomicAdd(&deg[(int)dst[e]], 1);
}

// ---------------------------------------------------------------------------
// dinv[i] = rsqrt(deg[i] + 1)   (+1 self-loop; always > 0)
// ---------------------------------------------------------------------------
__global__ __launch_bounds__(256) void GRACE_dinv(const int* __restrict__ deg,
                                                  float* __restrict__ dinv, int n) {
    int i = blockIdx.x * blockDim.x + threadIdx.x;
    if (i < n) dinv[i] = rsqrtf((float)deg[i] + 1.0f);
}

// ---------------------------------------------------------------------------
// One-block exclusive prefix scan of deg -> rowptr (N+1), cursor copy.
// 256 threads, each owns a contiguous chunk; Hillis-Steele over partials in LDS.
// ---------------------------------------------------------------------------
__global__ __launch_bounds__(SCAN_T) void GRACE_scan(const int* __restrict__ deg,
                                                     int* __restrict__ rowptr,
                                                     int* __restrict__ cursor, int n) {
    __shared__ int part[SCAN_T];
    const int t = threadIdx.x;
    const int chunk = (n + SCAN_T - 1) / SCAN_T;
    const int lo = t * chunk;
    const int hi = (lo + chunk < n) ? (lo + chunk) : n;
    int sum = 0;
    for (int i = lo; i < hi; ++i) sum += deg[i];
    part[t] = sum;
    __syncthreads();
    for (int off = 1; off < SCAN_T; off <<= 1) {
        int v = (t >= off) ? part[t - off] : 0;
        __syncthreads();
        part[t] += v;
        __syncthreads();
    }
    int run = (t == 0) ? 0 : part[t - 1];     // exclusive base for this chunk
    for (int i = lo; i < hi; ++i) {
        rowptr[i] = run;
        cursor[i] = run;
        run += deg[i];
    }
    if (t == SCAN_T - 1) rowptr[n] = run;     // == E
}

// ---------------------------------------------------------------------------
// CSR fill: place src index and precomputed norm into the dst row.
// ---------------------------------------------------------------------------
__global__ __launch_bounds__(256) void GRACE_fill(const long long* __restrict__ src,
                                                  const long long* __restrict__ dst,
                                                  const float* __restrict__ dinv,
                                                  int* __restrict__ cursor,
                                                  int* __restrict__ srcidx,
                                                  float* __restrict__ ew, int nedges) {
    int e = blockIdx.x * blockDim.x + threadIdx.x;
    if (e >= nedges) return;
    int s = (int)src[e];
    int d = (int)dst[e];
    int pos = atomicAdd(&cursor[d], 1);
    srcidx[pos] = s;
    ew[pos] = dinv[s] * dinv[d];
}

// ---------------------------------------------------------------------------
// Dense GEMM  H[nrows, fout] = X[nrows, K] @ W[K, fout]
// fp32 WMMA 16x16x4; one wave computes a 16(M) x 64(N) tile (4 accumulators
// reuse one A fragment). nrows%16==0, K%4==0, fout%64==0 hold here.
// ---------------------------------------------------------------------------
__global__ __launch_bounds__(256) void GRACE_gemm_wmma_f32(
    const float* __restrict__ X, const float* __restrict__ W,
    float* __restrict__ H, int nrows, int K, int fout) {
    const int gwave  = (blockIdx.x * blockDim.x + threadIdx.x) >> 5;
    const int lane   = threadIdx.x & 31;
    const int ntiles = fout >> 6;
    const int m0     = (gwave / ntiles) << 4;
    if (m0 >= nrows) return;                   // wave-uniform: EXEC stays all-1
    const int n0 = (gwave % ntiles) << 6;
    const int hi = lane >> 4;
    const int lo = lane & 15;

    const float* __restrict__ xrow = X + (size_t)(m0 + lo) * K;
    v8f acc0 = {}, acc1 = {}, acc2 = {}, acc3 = {};

    for (int k = 0; k < K; k += 4) {
        v2f a;
        a.x = xrow[k + 2 * hi];
        a.y = xrow[k + 2 * hi + 1];
        const float* __restrict__ wr0 = W + (size_t)(k + 2 * hi) * fout + n0 + lo;
        const float* __restrict__ wr1 = wr0 + fout;
        v2f b0; b0.x = wr0[0];  b0.y = wr1[0];
        v2f b1; b1.x = wr0[16]; b1.y = wr1[16];
        v2f b2; b2.x = wr0[32]; b2.y = wr1[32];
        v2f b3; b3.x = wr0[48]; b3.y = wr1[48];
        acc0 = __builtin_amdgcn_wmma_f32_16x16x4_f32(false, a, false, b0, (short)0, acc0, false, false);
        acc1 = __builtin_amdgcn_wmma_f32_16x16x4_f32(false, a, false, b1, (short)0, acc1, false, false);
        acc2 = __builtin_amdgcn_wmma_f32_16x16x4_f32(false, a, false, b2, (short)0, acc2, false, false);
        acc3 = __builtin_amdgcn_wmma_f32_16x16x4_f32(false, a, false, b3, (short)0, acc3, false, false);
    }

    float* hrow = H + (size_t)(m0 + 8 * hi) * fout + n0 + lo;
    #pragma unroll
    for (int v = 0; v < 8; ++v) {
        float* p = hrow + (size_t)v * fout;
        p[0]  = acc0[v];
        p[16] = acc1[v];
        p[32] = acc2[v];
        p[48] = acc3[v];
    }
}

// ---------------------------------------------------------------------------
// Gather-style aggregation, fused epilogue. One block per destination node:
//   out[d,c] = relu( sum_{e in row d} H[src[e],c]*w[e]
//                    + H[d,c]*dinv[d]^2 + bias[c] )
// blockDim = fout/2; each thread owns a float2 channel pair (b64 traffic).
// No atomics; each output element written exactly once.
// ---------------------------------------------------------------------------
__global__ __launch_bounds__(256) void GRACE_aggregate(
    const float* __restrict__ H, const int* __restrict__ rowptr,
    const int* __restrict__ srcidx, const float* __restrict__ ew,
    const float* __restrict__ dinv, const float* __restrict__ bias,
    float* __restrict__ out, int fout) {
    const int d = blockIdx.x;
    const int c = 2 * threadIdx.x;
    const int begin = rowptr[d];
    const int end   = rowptr[d + 1];
    const float sd = dinv[d];
    const float s2 = sd * sd;
    const size_t dbase = (size_t)d * fout;

    // self-loop + bias init
    const float2 hv = *(const float2*)(H + dbase + c);
    float a0 = hv.x * s2 + bias[c];
    float a1 = hv.y * s2 + bias[c + 1];

    for (int j = begin; j < end; ++j) {
        const int   s = srcidx[j];     // block-uniform -> scalar load
        const float w = ew[j];
        const float2 v = *(const float2*)(H + (size_t)s * fout + c);
        a0 += v.x * w;
        a1 += v.y * w;
    }
    float2 r;
    r.x = a0 > 0.0f ? a0 : 0.0f;
    r.y = a1 > 0.0f ? a1 : 0.0f;
    *(float2*)(out + dbase + c) = r;
}

// ---------------------------------------------------------------------------
// Launcher
// ---------------------------------------------------------------------------
extern "C" void kernel_launch(void* const* d_in, const int* in_sizes, int n_in,
                              void* d_out, int out_size, void* d_ws, size_t ws_size,
                              hipStream_t stream) {
    const float*     x   = (const float*)d_in[0];
    const float*     W1  = (const float*)d_in[1];
    const float*     b1  = (const float*)d_in[2];
    const float*     W2  = (const float*)d_in[3];
    const float*     b2  = (const float*)d_in[4];
    const long long* ei  = (const long long*)d_in[5];   // int64 per reference
    const long long* src = ei;
    const long long* dst = ei + EDGES;

    char*  ws     = (char*)d_ws;
    float* dinv   = (float*)(ws);                        // NODES f32
    int*   degcnt = (int*)(ws + (1 << 16));              // NODES i32
    int*   rowptr = (int*)(ws + (2 << 16));              // NODES+1 i32
    int*   cursor = (int*)(ws + (3 << 16));              // NODES i32
    int*   srcidx = (int*)(ws + (4 << 16));              // EDGES i32
    float* ew     = (float*)(ws + (4 << 16) + 4 * (size_t)EDGES);   // EDGES f32
    float* buf1   = (float*)(ws + (4 << 16) + 8 * (size_t)EDGES);   // NODES*HID
    float* buf2   = buf1 + (size_t)NODES * HID;                     // NODES*HID
    float* out    = (float*)d_out;

    // --- build normalization + CSR (graph shared by both layers) ---
    GRACE_zero_i32<<<(NODES + 255) / 256, 256, 0, stream>>>(degcnt, NODES);
    GRACE_count   <<<(EDGES + 255) / 256, 256, 0, stream>>>(dst, degcnt, EDGES);
    GRACE_dinv    <<<(NODES + 255) / 256, 256, 0, stream>>>(degcnt, dinv, NODES);
    GRACE_scan    <<<1, SCAN_T, 0, stream>>>(degcnt, rowptr, cursor, NODES);
    GRACE_fill    <<<(EDGES + 255) / 256, 256, 0, stream>>>(src, dst, dinv, cursor,
                                                            srcidx, ew, EDGES);

    // --- layer 1: h1 = x @ W1 ; relu(aggregate + self + b1) -> buf2 ---
    {
        int waves  = (NODES / 16) * (HID / 64);
        int blocks = (waves * 32 + 255) / 256;
        GRACE_gemm_wmma_f32<<<blocks, 256, 0, stream>>>(x, W1, buf1, NODES, INC, HID);
        GRACE_aggregate<<<NODES, HID / 2, 0, stream>>>(buf1, rowptr, srcidx, ew,
                                                       dinv, b1, buf2, HID);
    }

    // --- layer 2: h2 = buf2 @ W2 ; relu(aggregate + self + b2) -> d_out ---
    {
        int waves  = (NODES / 16) * (OUTC / 64);
        int blocks = (waves * 32 + 255) / 256;
        GRACE_gemm_wmma_f32<<<blocks, 256, 0, stream>>>(buf2, W2, buf1, NODES, HID, OUTC);
        GRACE_aggregate<<<NODES, OUTC / 2, 0, stream>>>(buf1, rowptr, srcidx, ew,
                                                        dinv, b2, out, OUTC);
    }
}